// GeneratingReconstructionLoss_34385508171853
// MI455X (gfx1250) — compile-verified
//
#include <hip/hip_runtime.h>
#include <hip/hip_bf16.h>
#include <math.h>

// Problem shape (from reference)
#define MAX_STEPS 16
#define BATCH     256
#define VOCAB     32000
#define ROWS      (MAX_STEPS * BATCH)   // 4096 softmax rows
#define V4        (VOCAB / 4)           // 8000 float4 per row
#define TPB       256
#define WAVES     (TPB / 32)            // wave32 on gfx1250

// Pipeline geometry: 8 chunks of 1000 float4 (16000 B); double buffer = 32 KB LDS
#define CHUNK_F4  1000
#define NCHUNK    (V4 / CHUNK_F4)       // 8
#define ITERS     4                     // ceil(CHUNK_F4/TPB) async b128 per wave per chunk

// ---------------- CDNA5 async global->LDS staging ----------------
typedef int v4i __attribute__((ext_vector_type(4)));
typedef __attribute__((address_space(1))) v4i gv4i;   // global (AS1)
typedef __attribute__((address_space(3))) v4i lv4i;   // LDS (AS3)

#if __has_builtin(__builtin_amdgcn_global_load_async_to_lds_b128)
#define HAVE_ASYNC_B128 1
#else
#define HAVE_ASYNC_B128 0
#warning "CDNA5 probe: async-to-LDS builtin NOT available; using VGPR staging fallback"
#endif

__device__ __forceinline__ void async_copy16(const float4* g, float4* l) {
#if HAVE_ASYNC_B128
  // (global src, LDS dst, imm offset, imm cpol)
  __builtin_amdgcn_global_load_async_to_lds_b128((gv4i*)g, (lv4i*)l, 0, 0);
#else
  *l = *g;
#endif
}

// Allow the just-issued next chunk (ITERS instr/wave) to stay in flight;
// in-order completion guarantees the previous chunk has landed.
__device__ __forceinline__ void async_wait_prev_chunk() {
#if __has_builtin(__builtin_amdgcn_s_wait_asynccnt)
  __builtin_amdgcn_s_wait_asynccnt(ITERS);
#else
  asm volatile("s_wait_asynccnt 4" ::: "memory");
#endif
}
__device__ __forceinline__ void async_wait_all() {
#if __has_builtin(__builtin_amdgcn_s_wait_asynccnt)
  __builtin_amdgcn_s_wait_asynccnt(0);
#else
  asm volatile("s_wait_asynccnt 0" ::: "memory");
#endif
}

// Log-sum-exp pair merge: (m,s) <- merge((m1,s1),(m2,s2))
__device__ __forceinline__ void lse_merge(float& m, float& s, float om, float os) {
  const float M = fmaxf(m, om);
  s = s * __expf(m - M) + os * __expf(om - M);
  m = M;
}

// ---------------- Kernel 1: one workgroup per (step,batch) row ----------------
// Double-buffered async staging: while chunk c is consumed out of LDS, chunk c+1
// streams in via global_load_async_to_lds_b128. Each thread consumes exactly the
// float4s it loaded, so only per-wave s_wait_asynccnt is needed (no barriers in
// the hot loop). Online (running max + lazy rescale) softmax -> one HBM pass.
__global__ __launch_bounds__(TPB) void ponder_ce_rows(
    const float* __restrict__ p,        // (MAX_STEPS, BATCH)
    const float* __restrict__ y_pred,   // (MAX_STEPS, BATCH, VOCAB)
    const int*   __restrict__ y_true,   // (BATCH,)
    float*       __restrict__ partial)  // (ROWS,) scratch
{
  __shared__ __align__(16) float4 buf[2][CHUNK_F4];   // 32000 B
  __shared__ float redm[WAVES], reds[WAVES];

  const int r   = blockIdx.x;           // r = n*BATCH + b
  const int tid = threadIdx.x;
  const float*  row  = y_pred + (size_t)r * VOCAB;
  const float4* row4 = (const float4*)row;

  // ---- prologue: stage chunk 0 ----
#pragma unroll
  for (int j = 0; j < ITERS; ++j) {
    const int i = tid + j * TPB;
    if (i < CHUNK_F4) async_copy16(row4 + i, &buf[0][i]);
  }

  float m = -__builtin_inff();
  float s = 0.f;

  for (int c = 0; c < NCHUNK; ++c) {
    const int cb = c & 1;
    if (c + 1 < NCHUNK) {
      const float4* src = row4 + (size_t)(c + 1) * CHUNK_F4;
#pragma unroll
      for (int j = 0; j < ITERS; ++j) {
        const int i = tid + j * TPB;
        if (i < CHUNK_F4) async_copy16(src + i, &buf[cb ^ 1][i]);
      }
      async_wait_prev_chunk();          // chunk c landed; c+1 still in flight
    } else {
      async_wait_all();                 // last chunk: drain
    }

    // ---- consume chunk c (per-thread online softmax) ----
#pragma unroll
    for (int j = 0; j < ITERS; ++j) {
      const int i = tid + j * TPB;
      if (i < CHUNK_F4) {
        const float4 v  = buf[cb][i];
        const float  cm = fmaxf(fmaxf(v.x, v.y), fmaxf(v.z, v.w));
        if (cm > m) { s *= __expf(m - cm); m = cm; }   // lazy rescale (rare)
        s += __expf(v.x - m) + __expf(v.y - m) +
             __expf(v.z - m) + __expf(v.w - m);
      }
    }
  }

  // ---- wave32 merge of (m, s) ----
#pragma unroll
  for (int off = 16; off >= 1; off >>= 1) {
    const float om = __shfl_xor(m, off);
    const float os = __shfl_xor(s, off);
    lse_merge(m, s, om, os);
  }
  if ((tid & 31) == 0) { redm[tid >> 5] = m; reds[tid >> 5] = s; }
  __syncthreads();

  if (tid == 0) {
    float M = redm[0], S = reds[0];
#pragma unroll
    for (int w = 1; w < WAVES; ++w) lse_merge(M, S, redm[w], reds[w]);
    const int   b  = r & (BATCH - 1);
    const float xt = row[y_true[b]];              // gather true-class logit
    const float ce = M + logf(S) - xt;            // lse - x_t
    partial[r] = p[r] * ce;                       // p flat index == r
  }
}

// ---------------- Kernel 2: deterministic fixed-order final reduction ----------------
__global__ __launch_bounds__(TPB) void ponder_ce_reduce(
    const float* __restrict__ partial, float* __restrict__ out)
{
  __shared__ float sm[TPB];
  const int tid = threadIdx.x;
  float s = 0.f;
  for (int i = tid; i < ROWS; i += TPB) s += partial[i];  // 16 adds, fixed order
  sm[tid] = s;
  __syncthreads();
#pragma unroll
  for (int off = TPB / 2; off > 0; off >>= 1) {
    if (tid < off) sm[tid] += sm[tid + off];
    __syncthreads();
  }
  if (tid == 0) out[0] = sm[0] * (1.0f / BATCH);
}

// ---------------- Host-side launcher ----------------
extern "C" void kernel_launch(void* const* d_in, const int* in_sizes, int n_in,
                              void* d_out, int out_size, void* d_ws, size_t ws_size,
                              hipStream_t stream) {
  const float* p      = (const float*)d_in[0];
  const float* y_pred = (const float*)d_in[1];
  const int*   y_true = (const int*)d_in[2];
  float* partial = (float*)d_ws;          // ROWS * 4 B = 16 KB scratch
  float* out     = (float*)d_out;

  ponder_ce_rows<<<ROWS, TPB, 0, stream>>>(p, y_pred, y_true, partial);
  ponder_ce_reduce<<<1, TPB, 0, stream>>>(partial, out);
}